// S2S2SModel_16569983828315
// MI455X (gfx1250) — compile-verified
//
#include <hip/hip_runtime.h>

// ---------------------------------------------------------------------------
// Persistent-grid fp32 LSTM seq2seq for MI455X (gfx1250), wave32 + WMMA f32.
//
// B=256, H=256, 4H=1024.  Grid = 16 WGs x 512 threads (16 waves).
//   WG  g  owns hidden columns [16g, 16g+16)  (gate rows {q*256+16g+n}).
//   Wave w owns batch rows     [16w, 16w+16).
// Each wave's 16x16 cell-state tile stays in a v8f register accumulator for
// all 456 timesteps.  Per step: gates tile = h x Whh^T + x_t x Wih^T + b via
// V_WMMA_F32_16X16X4_F32 (exact fp32), then lane-local sigmoid/tanh update,
// h tile written to a ping-pong global buffer, one grid-wide barrier.
//
// All global traffic is forced into addrspace(1) so the A-stream uses
// global_load_b64 (LOADcnt only) and never contends with the LDS B-stream
// (ds_load, DScnt) the way flat_load (LOADcnt+DScnt) would.
// ---------------------------------------------------------------------------

typedef __attribute__((ext_vector_type(2))) float v2f;
typedef __attribute__((ext_vector_type(8))) float v8f;

#define AS_GLOBAL __attribute__((address_space(1)))
typedef const AS_GLOBAL float* gcfloat_p;
typedef AS_GLOBAL float* gfloat_p;
typedef const AS_GLOBAL v2f* gcv2f_p;

#define NWG     16
#define WGSIZE  512
#define HH      256
#define BB      256
#define WHH_S   260   // LDS row stride (floats): 260 % 64 == 4 -> no bank conflicts
#define WIH_S   68    // 68 % 64 == 4

__device__ __forceinline__ float sigmoidf_(float x) {
  return 1.0f / (1.0f + __expf(-x));
}

// Device-wide generation barrier across the 16 persistent workgroups.
__device__ __forceinline__ void grid_barrier(unsigned* cnt, unsigned* gen) {
  __threadfence();            // publish this thread's h stores device-wide
  __syncthreads();
  if (threadIdx.x == 0) {
    unsigned g = __hip_atomic_load(gen, __ATOMIC_RELAXED, __HIP_MEMORY_SCOPE_AGENT);
    unsigned arrived =
        __hip_atomic_fetch_add(cnt, 1u, __ATOMIC_ACQ_REL, __HIP_MEMORY_SCOPE_AGENT);
    if (arrived == NWG - 1u) {
      __hip_atomic_store(cnt, 0u, __ATOMIC_RELAXED, __HIP_MEMORY_SCOPE_AGENT);
      __hip_atomic_fetch_add(gen, 1u, __ATOMIC_RELEASE, __HIP_MEMORY_SCOPE_AGENT);
    } else {
      while (__hip_atomic_load(gen, __ATOMIC_ACQUIRE, __HIP_MEMORY_SCOPE_AGENT) == g) {
        __builtin_amdgcn_s_sleep(2);
      }
    }
  }
  __syncthreads();
}

// Stage this WG's weight slice into LDS.
// lds_whh: [64 rows (gate-major: q*16+n)][WHH_S]  (row n of gate q = Whh row q*256+16g+n)
// lds_wih: [64 rows][WIH_S], lds_bias: [64]
__device__ __forceinline__ void load_weights(gcfloat_p Wih, gcfloat_p Whh,
                                             gcfloat_p bias, int D, float* lds_whh,
                                             float* lds_wih, float* lds_bias) {
  const int g = blockIdx.x;
  for (int idx = threadIdx.x; idx < 64 * HH; idx += WGSIZE) {   // 16384/512: uniform
    int r = idx >> 8;
    int k = idx & 255;
    int q = r >> 4, n = r & 15;
    lds_whh[r * WHH_S + k] = Whh[(q * HH + 16 * g + n) * HH + k];
  }
  for (int idx = threadIdx.x; idx < 64 * D; idx += WGSIZE) {    // uniform trip counts
    int r = idx / D;
    int k = idx - r * D;
    int q = r >> 4, n = r & 15;
    lds_wih[r * WIH_S + k] = Wih[(q * HH + 16 * g + n) * D + k];
  }
  if (threadIdx.x < 64) {
    int q = threadIdx.x >> 4, n = threadIdx.x & 15;
    lds_bias[threadIdx.x] = bias[q * HH + 16 * g + n];
  }
  __syncthreads();
}

// One LSTM step for this wave's 16x16 tile.
__device__ __forceinline__ void do_step(gcfloat_p hcur, gfloat_p hnxt, gcfloat_p xbase,
                                        int x_row_stride, int x_off, int D,
                                        const float* __restrict__ lds_whh,
                                        const float* __restrict__ lds_wih,
                                        const float* __restrict__ lds_bias,
                                        int rowBase, int colBase, int m, int kh,
                                        v8f* c_acc) {
  v8f acc0 = {};  // i gate, 16x16 f32 accumulator
  v8f acc1 = {};  // f
  v8f acc2 = {};  // g
  v8f acc3 = {};  // o

  // Recurrent part: K = 256 over h, four gate tiles share each A fragment.
  gcfloat_p hrow = hcur + (rowBase + m) * HH;
  for (int k0 = 0; k0 < HH; k0 += 4) {
    v2f a = *(gcv2f_p)(hrow + k0 + 2 * kh);                      // A[m][k0+2kh..+1]
    const float* wb = lds_whh + k0 + 2 * kh;                     // B[k][n], n==m lane id
    v2f b0 = *(const v2f*)(wb + (0 * 16 + m) * WHH_S);
    v2f b1 = *(const v2f*)(wb + (1 * 16 + m) * WHH_S);
    v2f b2 = *(const v2f*)(wb + (2 * 16 + m) * WHH_S);
    v2f b3 = *(const v2f*)(wb + (3 * 16 + m) * WHH_S);
    acc0 = __builtin_amdgcn_wmma_f32_16x16x4_f32(false, a, false, b0, (short)0, acc0, false, false);
    acc1 = __builtin_amdgcn_wmma_f32_16x16x4_f32(false, a, false, b1, (short)0, acc1, false, false);
    acc2 = __builtin_amdgcn_wmma_f32_16x16x4_f32(false, a, false, b2, (short)0, acc2, false, false);
    acc3 = __builtin_amdgcn_wmma_f32_16x16x4_f32(false, a, false, b3, (short)0, acc3, false, false);
  }

  // Input part: K = D (64/32/16).
  gcfloat_p xrow = xbase + (rowBase + m) * x_row_stride + x_off;
  for (int k0 = 0; k0 < D; k0 += 4) {
    v2f a = *(gcv2f_p)(xrow + k0 + 2 * kh);
    const float* wb = lds_wih + k0 + 2 * kh;
    v2f b0 = *(const v2f*)(wb + (0 * 16 + m) * WIH_S);
    v2f b1 = *(const v2f*)(wb + (1 * 16 + m) * WIH_S);
    v2f b2 = *(const v2f*)(wb + (2 * 16 + m) * WIH_S);
    v2f b3 = *(const v2f*)(wb + (3 * 16 + m) * WIH_S);
    acc0 = __builtin_amdgcn_wmma_f32_16x16x4_f32(false, a, false, b0, (short)0, acc0, false, false);
    acc1 = __builtin_amdgcn_wmma_f32_16x16x4_f32(false, a, false, b1, (short)0, acc1, false, false);
    acc2 = __builtin_amdgcn_wmma_f32_16x16x4_f32(false, a, false, b2, (short)0, acc2, false, false);
    acc3 = __builtin_amdgcn_wmma_f32_16x16x4_f32(false, a, false, b3, (short)0, acc3, false, false);
  }

  // Lane-local cell update.  D-layout: lane holds column n=m; VGPR e holds
  // batch row (e + 8*kh).  Bias is per gate column.
  const float bi = lds_bias[0 * 16 + m];
  const float bf = lds_bias[1 * 16 + m];
  const float bg = lds_bias[2 * 16 + m];
  const float bo = lds_bias[3 * 16 + m];
  v8f cc = *c_acc;
  gfloat_p hdst = hnxt + colBase + m;
#pragma unroll
  for (int e = 0; e < 8; e++) {
    float ig = sigmoidf_(acc0[e] + bi);
    float fg = sigmoidf_(acc1[e] + bf);
    float gg = tanhf(acc2[e] + bg);
    float og = sigmoidf_(acc3[e] + bo);
    float cv = fg * cc[e] + ig * gg;
    cc[e] = cv;
    hdst[(rowBase + e + 8 * kh) * HH] = og * tanhf(cv);
  }
  *c_acc = cc;
}

__global__ __launch_bounds__(WGSIZE) void lstm_s2s2s_kernel(
    const float* __restrict__ x_d, const float* __restrict__ x_f,
    const float* __restrict__ x_ff, const float* __restrict__ enc_Wih,
    const float* __restrict__ enc_Whh, const float* __restrict__ enc_b,
    const float* __restrict__ decf_Wih, const float* __restrict__ decf_Whh,
    const float* __restrict__ decf_b, const float* __restrict__ decu_Wih,
    const float* __restrict__ decu_Whh, const float* __restrict__ decu_b,
    const float* __restrict__ head_W, const float* __restrict__ head_b,
    float* __restrict__ out, unsigned* __restrict__ bar,
    float* __restrict__ hbase /* 2 x BB*HH ping-pong */) {
  extern __shared__ float smem[];
  float* lds_whh  = smem;
  float* lds_wih  = smem + 64 * WHH_S;
  float* lds_bias = lds_wih + 64 * WIH_S;

  unsigned* cnt = bar;
  unsigned* gen = bar + 1;

  const int wave    = threadIdx.x >> 5;
  const int lane    = threadIdx.x & 31;
  const int m       = lane & 15;
  const int kh      = lane >> 4;
  const int rowBase = wave * 16;         // batch rows owned by this wave
  const int colBase = blockIdx.x * 16;   // hidden cols owned by this WG

  // Force global address space everywhere (global_load/_store, never flat).
  gfloat_p  hg    = (gfloat_p)hbase;     // [2][BB*HH]; slice 0 zeroed by memsetAsync
  gcfloat_p xdg   = (gcfloat_p)x_d;
  gcfloat_p xfg   = (gcfloat_p)x_f;
  gcfloat_p xffg  = (gcfloat_p)x_ff;
  gcfloat_p hWg   = (gcfloat_p)head_W;
  gfloat_p  outg  = (gfloat_p)out;

  v8f c_acc = {};                        // cell state tile lives in registers
  int cur = 0;

  // ---- Encoder: 365 steps over x_d [256][365][64] ----
  load_weights((gcfloat_p)enc_Wih, (gcfloat_p)enc_Whh, (gcfloat_p)enc_b, 64,
               lds_whh, lds_wih, lds_bias);
  for (int t = 0; t < 365; t++) {
    do_step(hg + cur * (BB * HH), hg + (cur ^ 1) * (BB * HH), xdg, 365 * 64, t * 64,
            64, lds_whh, lds_wih, lds_bias, rowBase, colBase, m, kh, &c_acc);
    grid_barrier(cnt, gen);
    cur ^= 1;
  }

  // ---- Forecast decoder: 15 steps over x_f [256][15][32] (head overwritten) ----
  load_weights((gcfloat_p)decf_Wih, (gcfloat_p)decf_Whh, (gcfloat_p)decf_b, 32,
               lds_whh, lds_wih, lds_bias);
  for (int t = 0; t < 15; t++) {
    do_step(hg + cur * (BB * HH), hg + (cur ^ 1) * (BB * HH), xfg, 15 * 32, t * 32,
            32, lds_whh, lds_wih, lds_bias, rowBase, colBase, m, kh, &c_acc);
    grid_barrier(cnt, gen);
    cur ^= 1;
  }

  // ---- Future decoder: 76 steps over x_ff [256][76][16] + linear head ----
  load_weights((gcfloat_p)decu_Wih, (gcfloat_p)decu_Whh, (gcfloat_p)decu_b, 16,
               lds_whh, lds_wih, lds_bias);
  const float hb0 = head_b[0];
  for (int t = 0; t < 76; t++) {
    do_step(hg + cur * (BB * HH), hg + (cur ^ 1) * (BB * HH), xffg, 76 * 16, t * 16,
            16, lds_whh, lds_wih, lds_bias, rowBase, colBase, m, kh, &c_acc);
    grid_barrier(cnt, gen);
    if (blockIdx.x == 0) {
      // WG0 computes out[:, t] = h_t @ head_W + head_b while others start t+1;
      // buffer reuse is protected by the next grid barrier.
      gcfloat_p hnow = hg + (cur ^ 1) * (BB * HH);
      for (int b = threadIdx.x; b < BB; b += WGSIZE) {
        gcfloat_p hr = hnow + b * HH;
        float s = hb0;
        for (int k = 0; k < HH; k++) s = fmaf(hr[k], hWg[k], s);
        outg[b * 91 + t] = s;   // cols 76..90 stay zero (memset) per reference
      }
    }
    cur ^= 1;
  }
}

extern "C" void kernel_launch(void* const* d_in, const int* in_sizes, int n_in,
                              void* d_out, int out_size, void* d_ws, size_t ws_size,
                              hipStream_t stream) {
  (void)in_sizes; (void)n_in; (void)ws_size;
  const float* x_d      = (const float*)d_in[0];
  const float* x_f      = (const float*)d_in[1];
  const float* x_ff     = (const float*)d_in[2];
  const float* enc_Wih  = (const float*)d_in[3];
  const float* enc_Whh  = (const float*)d_in[4];
  const float* enc_b    = (const float*)d_in[5];
  const float* decf_Wih = (const float*)d_in[6];
  const float* decf_Whh = (const float*)d_in[7];
  const float* decf_b   = (const float*)d_in[8];
  const float* decu_Wih = (const float*)d_in[9];
  const float* decu_Whh = (const float*)d_in[10];
  const float* decu_b   = (const float*)d_in[11];
  const float* head_W   = (const float*)d_in[12];
  const float* head_b   = (const float*)d_in[13];
  float* out = (float*)d_out;

  unsigned char* ws = (unsigned char*)d_ws;
  unsigned* bar = (unsigned*)ws;               // {cnt, gen}
  float* hbase  = (float*)(ws + 256);          // 2 x 256KB ping-pong h buffers

  // Zero barrier state + h0 each call (ws is poisoned, not re-poisoned).
  hipMemsetAsync(d_ws, 0, 256 + (size_t)BB * HH * 4, stream);
  // Output columns [76,91) must be zero; head overwrites [0,76).
  hipMemsetAsync(d_out, 0, (size_t)out_size * sizeof(float), stream);

  size_t smem_bytes = (size_t)(64 * WHH_S + 64 * WIH_S + 64) * sizeof(float);
  hipLaunchKernelGGL(lstm_s2s2s_kernel, dim3(NWG), dim3(WGSIZE), smem_bytes, stream,
                     x_d, x_f, x_ff, enc_Wih, enc_Whh, enc_b, decf_Wih, decf_Whh,
                     decf_b, decu_Wih, decu_Whh, decu_b, head_W, head_b, out, bar,
                     hbase);
}